// WriteHead_36730560315416
// MI455X (gfx1250) — compile-verified
//
#include <hip/hip_runtime.h>
#include <math.h>

// Dims from reference
#define H 1024
#define N 512
#define M 128
#define NM (N * M)
#define EPS 1e-8f

typedef float v2f __attribute__((ext_vector_type(2)));
typedef float v4f __attribute__((ext_vector_type(4)));
typedef float v8f __attribute__((ext_vector_type(8)));

__device__ __forceinline__ float waveReduceSum(float v) {
    // wave32 butterfly reduction
    #pragma unroll
    for (int off = 16; off >= 1; off >>= 1)
        v += __shfl_xor(v, off, 32);
    return v;
}

// ---------------------------------------------------------------------------
// K1: small projections.  One wave per output row.
// jobs 0..127   -> k[j] = Wk[j,:] . x + bk[j]
// jobs 128..255 -> a[j] = Wa[j-128,:] . x + ba[j-128]
// job 256       -> beta = softplus(Wbeta . x + bbeta)
// job 257       -> g    = sigmoid (Wg    . x + bg)
// ---------------------------------------------------------------------------
__global__ void proj_kernel(const float* __restrict__ x,
                            const float* __restrict__ Wk,    const float* __restrict__ bk,
                            const float* __restrict__ Wa,    const float* __restrict__ ba,
                            const float* __restrict__ Wbeta, const float* __restrict__ bbeta,
                            const float* __restrict__ Wg,    const float* __restrict__ bg,
                            float* __restrict__ ws_k, float* __restrict__ ws_a,
                            float* __restrict__ ws_scal) {
    int wave = blockIdx.x * (blockDim.x >> 5) + (threadIdx.x >> 5);
    int lane = threadIdx.x & 31;
    if (wave >= 258) return;

    const float* row;
    float bias;
    if (wave < 128)        { row = Wk + wave * H;          bias = bk[wave]; }
    else if (wave < 256)   { row = Wa + (wave - 128) * H;  bias = ba[wave - 128]; }
    else if (wave == 256)  { row = Wbeta;                  bias = bbeta[0]; }
    else                   { row = Wg;                     bias = bg[0]; }

    float acc = 0.0f;
    #pragma unroll 4
    for (int t = lane; t < H; t += 32)
        acc = fmaf(row[t], x[t], acc);
    acc = waveReduceSum(acc);

    if (lane == 0) {
        float r = acc + bias;
        if (wave < 128)       ws_k[wave] = r;
        else if (wave < 256)  ws_a[wave - 128] = r;
        else if (wave == 256) ws_scal[0] = logf(1.0f + expf(r));     // softplus
        else                  ws_scal[1] = 1.0f / (1.0f + expf(-r)); // sigmoid
    }
}

// ---------------------------------------------------------------------------
// K2: cosine-similarity logits.  One wave per memory row (512 waves).
// z[i] = beta * (mem[i,:].k) / (max(||mem[i]||,eps) * max(||k||,eps))
// ---------------------------------------------------------------------------
__global__ void cos_kernel(const float* __restrict__ mem,
                           const float* __restrict__ ws_k,
                           const float* __restrict__ ws_scal,
                           float* __restrict__ ws_z) {
    int wave = blockIdx.x * (blockDim.x >> 5) + (threadIdx.x >> 5); // 0..511
    int lane = threadIdx.x & 31;
    const float* row = mem + wave * M;

    float d = 0.0f, mm = 0.0f, kk = 0.0f;
    #pragma unroll
    for (int t = lane; t < M; t += 32) {
        float mv = row[t];
        float kv = ws_k[t];
        d  = fmaf(mv, kv, d);
        mm = fmaf(mv, mv, mm);
        kk = fmaf(kv, kv, kk);
    }
    d  = waveReduceSum(d);
    mm = waveReduceSum(mm);
    kk = waveReduceSum(kk);

    if (lane == 0) {
        float mn = fmaxf(sqrtf(mm), EPS);
        float kn = fmaxf(sqrtf(kk), EPS);
        ws_z[wave] = ws_scal[0] * (d / (mn * kn));
    }
}

// ---------------------------------------------------------------------------
// K3: softmax over 512 logits + gated interpolation with previous_state.
// Single block of 512 threads.  state -> d_out[128 .. 640)
// ---------------------------------------------------------------------------
__global__ void softmax_state_kernel(const float* __restrict__ ws_z,
                                     const float* __restrict__ ws_scal,
                                     const float* __restrict__ prev,
                                     float* __restrict__ state_out) {
    __shared__ float red[N];
    int t = threadIdx.x; // 0..511
    float z = ws_z[t];

    red[t] = z;
    __syncthreads();
    for (int s = N / 2; s >= 1; s >>= 1) {
        if (t < s) red[t] = fmaxf(red[t], red[t + s]);
        __syncthreads();
    }
    float mx = red[0];
    __syncthreads();

    float ez = expf(z - mx);
    red[t] = ez;
    __syncthreads();
    for (int s = N / 2; s >= 1; s >>= 1) {
        if (t < s) red[t] += red[t + s];
        __syncthreads();
    }
    float sum = red[0];

    float w = ez / sum;
    float g = ws_scal[1];
    state_out[t] = g * w + (1.0f - g) * prev[t];
}

// ---------------------------------------------------------------------------
// K4: read = state @ mem  -> d_out[0 .. 128)
// ---------------------------------------------------------------------------
__global__ void read_kernel(const float* __restrict__ state,
                            const float* __restrict__ mem,
                            float* __restrict__ read_out) {
    int j = threadIdx.x; // 0..127
    float acc = 0.0f;
    #pragma unroll 4
    for (int i = 0; i < N; ++i)
        acc = fmaf(state[i], mem[i * M + j], acc);
    read_out[j] = acc;
}

// ---------------------------------------------------------------------------
// K5: bandwidth-dominant kernel.  e = sigmoid(We @ x + be) fused with the
// memory update, using V_WMMA_F32_16X16X4_F32.
//
// Each wave owns a 16-row tile of We (65536 rows -> 4096 waves).
// A is streamed with b128 loads: per lane, float4 at row*H + kb + 2*koff
// (koff = lane<16 ? 0 : 2).  Two WMMAs consume the xy / zw halves:
//   WMMA#1 contracts physical k in {kb, kb+1, kb+4, kb+5}
//   WMMA#2 contracts physical k in {kb+2, kb+3, kb+6, kb+7}
// with matching broadcast-x B fragments from LDS, so kb..kb+7 is covered
// exactly once per iteration (K summation order is commutative).
// After the loop, column N=0 of D sits in lane 0 (M=0..7) and lane 16
// (M=8..15) across the 8 accumulator VGPRs.
// ---------------------------------------------------------------------------
__global__ void __launch_bounds__(256)
erase_write_kernel(const float* __restrict__ x,
                   const float* __restrict__ We, const float* __restrict__ be,
                   const float* __restrict__ mem,
                   const float* __restrict__ state,
                   const float* __restrict__ a_vec,
                   float* __restrict__ newmem_out) {
    __shared__ float xs[H];
    for (int t = threadIdx.x; t < H; t += blockDim.x) xs[t] = x[t];
    __syncthreads();

    int wave = blockIdx.x * (blockDim.x >> 5) + (threadIdx.x >> 5); // 0..4095 exactly
    int lane = threadIdx.x & 31;
    int rowBase = wave * 16;
    int mrow = lane & 15;
    int koff2 = (lane < 16) ? 0 : 4;   // 2*koff

    const float* Aptr = We + (size_t)(rowBase + mrow) * H + koff2;
    const float* Bptr = xs + koff2;

    v8f acc = {};
    for (int kb = 0; kb < H; kb += 8) {
        v4f av = *(const v4f*)(Aptr + kb);          // global b128 (16B aligned)
        v2f b0 = *(const v2f*)(Bptr + kb);          // LDS b64
        v2f b1 = *(const v2f*)(Bptr + kb + 2);      // LDS b64
        v2f a0 = __builtin_shufflevector(av, av, 0, 1);
        v2f a1 = __builtin_shufflevector(av, av, 2, 3);
        acc = __builtin_amdgcn_wmma_f32_16x16x4_f32(
            false, a0, false, b0, (short)0, acc, false, false);
        acc = __builtin_amdgcn_wmma_f32_16x16x4_f32(
            false, a1, false, b1, (short)0, acc, false, false);
    }

    // Column N=0 holders: lane 0 -> M=0..7, lane 16 -> M=8..15.
    // Each active lane touches 8 consecutive flat indices -> b128-mergeable.
    if (mrow == 0) {
        int mbase = (lane == 0) ? 0 : 8;
        #pragma unroll
        for (int r = 0; r < 8; ++r) {
            int idx = rowBase + mbase + r;      // flat (i*M + j)
            float dot = acc[r] + be[idx];
            float e = 1.0f / (1.0f + expf(-dot));
            int i = idx >> 7;          // / M
            int j = idx & (M - 1);     // % M
            float st = state[i];
            newmem_out[idx] = mem[idx] * (1.0f - st * e) + st * a_vec[j];
        }
    }
}

// ---------------------------------------------------------------------------
extern "C" void kernel_launch(void* const* d_in, const int* in_sizes, int n_in,
                              void* d_out, int out_size, void* d_ws, size_t ws_size,
                              hipStream_t stream) {
    (void)in_sizes; (void)n_in; (void)out_size; (void)ws_size;
    const float* x     = (const float*)d_in[0];
    const float* prev  = (const float*)d_in[1];
    const float* mem   = (const float*)d_in[2];
    const float* Wk    = (const float*)d_in[3];
    const float* bk    = (const float*)d_in[4];
    const float* Wbeta = (const float*)d_in[5];
    const float* bbeta = (const float*)d_in[6];
    const float* Wg    = (const float*)d_in[7];
    const float* bg    = (const float*)d_in[8];
    // d_in[9..12] = Ws, bs, Wgamma, bgamma: computed-but-unused in reference; skipped.
    const float* We    = (const float*)d_in[13];
    const float* be    = (const float*)d_in[14];
    const float* Wa    = (const float*)d_in[15];
    const float* ba    = (const float*)d_in[16];

    float* out  = (float*)d_out;      // [0,128) read | [128,640) state | [640,66176) new_memory
    float* ws   = (float*)d_ws;
    float* ws_k = ws;                 // 128
    float* ws_a = ws + 128;           // 128
    float* ws_s = ws + 256;           // [0]=beta, [1]=g
    float* ws_z = ws + 320;           // 512 logits

    proj_kernel<<<dim3(65), dim3(128), 0, stream>>>(x, Wk, bk, Wa, ba,
                                                    Wbeta, bbeta, Wg, bg,
                                                    ws_k, ws_a, ws_s);
    cos_kernel<<<dim3(64), dim3(256), 0, stream>>>(mem, ws_k, ws_s, ws_z);
    softmax_state_kernel<<<dim3(1), dim3(512), 0, stream>>>(ws_z, ws_s, prev, out + 128);
    read_kernel<<<dim3(1), dim3(128), 0, stream>>>(out + 128, mem, out);
    erase_write_kernel<<<dim3(512), dim3(256), 0, stream>>>(x, We, be, mem,
                                                            out + 128, ws_a, out + 640);
}